// FP_86045374808178
// MI455X (gfx1250) — compile-verified
//
#include <hip/hip_runtime.h>
#include <hip/hip_bf16.h>

#define VIEWS 180
#define NBATCH 2
#define IMG 512

typedef __attribute__((ext_vector_type(2))) float v2f;
typedef __attribute__((ext_vector_type(8))) float v8f;

// One wave handles one view's 16 consecutive output columns for BOTH batch
// images.  h is consumed 4 rows per step via V_WMMA_F32_16X16X4_F32 with a
// constant A = 1/H, so D accumulates the column means directly.
__global__ __launch_bounds__(256) void radon_fp_wmma_kernel(
    const float* __restrict__ x,   // (2, 1, 512, 512)
    float* __restrict__ out)       // (2, 180, 512)
{
    const int lane  = threadIdx.x & 31;
    const int wave  = threadIdx.x >> 5;
    const int v     = blockIdx.x;                    // view index
    const int w0    = blockIdx.y * 128 + wave * 16;  // first output column
    const int n     = lane & 15;                     // column within wave tile
    const int kbase = (lane >> 4) << 1;              // 0 for lanes 0-15, 2 for 16-31
    const int w     = w0 + n;

    // thetas = linspace(0, pi, 180, endpoint=False); cos/sin in double, cast f32
    const double PI = 3.14159265358979323846;
    const double th = (double)v * (PI / (double)VIEWS);
    const float c = (float)cos(th);
    const float s = (float)sin(th);

    // align_corners=False base coords
    const float X0 = (2.0f * (float)w + 1.0f) * (1.0f / (float)IMG) - 1.0f;

    // this lane's two h rows per step: h = 4*step + (kbase + j), j in {0,1}
    float Y0 = (2.0f * (float)(kbase + 0) + 1.0f) * (1.0f / (float)IMG) - 1.0f;
    float Y1 = (2.0f * (float)(kbase + 1) + 1.0f) * (1.0f / (float)IMG) - 1.0f;
    const float dY = 8.0f / (float)IMG;  // exact in f32; increments stay exact

    const float invH = 1.0f / (float)IMG;
    v2f a; a.x = invH; a.y = invH;       // A(16x4) all 1/H -> D rows = column means

    v8f acc0 = {};                       // batch 0 accumulator (C/D chained)
    v8f acc1 = {};                       // batch 1 accumulator
    const float* __restrict__ xb1 = x + IMG * IMG;  // batch 1 image (C==1)

    for (int step = 0; step < IMG / 4; ++step) {
        float vals0[2], vals1[2];
#pragma unroll
        for (int j = 0; j < 2; ++j) {
            const float Yj = j ? Y1 : Y0;
            // rotate
            float gx = c * X0 - s * Yj;
            float gy = s * X0 + c * Yj;
            // torch.clamp(grid, -1, 1)
            gx = fminf(fmaxf(gx, -1.0f), 1.0f);
            gy = fminf(fmaxf(gy, -1.0f), 1.0f);
            // unnormalize (align_corners=False) + round-half-to-even
            float fx = ((gx + 1.0f) * (float)IMG - 1.0f) * 0.5f;
            float fy = ((gy + 1.0f) * (float)IMG - 1.0f) * 0.5f;
            fx = __builtin_rintf(fx);    // v_rndne_f32
            fy = __builtin_rintf(fy);
            const bool valid = (fx >= 0.0f) & (fx <= (float)(IMG - 1)) &
                               (fy >= 0.0f) & (fy <= (float)(IMG - 1));
            // clamp so the (unconditional) load address is always in-bounds
            const int ix = (int)fminf(fmaxf(fx, 0.0f), (float)(IMG - 1));
            const int iy = (int)fminf(fmaxf(fy, 0.0f), (float)(IMG - 1));
            const int off = iy * IMG + ix;
            const float p0 = x[off];     // L2-resident gather, batch 0
            const float p1 = xb1[off];   // same index reused for batch 1
            vals0[j] = valid ? p0 : 0.0f;  // zeros padding via cndmask, no branch
            vals1[j] = valid ? p1 : 0.0f;
        }
        v2f b0; b0.x = vals0[0]; b0.y = vals0[1];
        v2f b1; b1.x = vals1[0]; b1.y = vals1[1];
        // D = A*B + C ; 8-arg form: (neg_a, A, neg_b, B, c_mod, C, reuse_a, reuse_b)
        acc0 = __builtin_amdgcn_wmma_f32_16x16x4_f32(false, a, false, b0,
                                                     (short)0, acc0, false, false);
        acc1 = __builtin_amdgcn_wmma_f32_16x16x4_f32(false, a, false, b1,
                                                     (short)0, acc1, false, false);
        Y0 += dY;
        Y1 += dY;
    }

    // D rows are identical (A constant): VGPR0 lanes 0-15 hold proj[w0+lane].
    if (lane < 16) {
        out[(0 * VIEWS + v) * IMG + w] = acc0[0];
        out[(1 * VIEWS + v) * IMG + w] = acc1[0];
    }
}

extern "C" void kernel_launch(void* const* d_in, const int* in_sizes, int n_in,
                              void* d_out, int out_size, void* d_ws, size_t ws_size,
                              hipStream_t stream) {
    (void)in_sizes; (void)n_in; (void)d_ws; (void)ws_size; (void)out_size;
    const float* x = (const float*)d_in[0];
    float* out = (float*)d_out;
    dim3 grid(VIEWS, IMG / 128, 1);   // 180 x 4 blocks
    dim3 block(256, 1, 1);            // 8 waves; each wave -> 16 columns
    radon_fp_wmma_kernel<<<grid, block, 0, stream>>>(x, out);
}